// Net1_47639777247624
// MI455X (gfx1250) — compile-verified
//
#include <hip/hip_runtime.h>

typedef __attribute__((ext_vector_type(2))) float v2f;
typedef __attribute__((ext_vector_type(8))) float v8f;

#define NSITES 100000
#define NYEARS 20
#define NE     (NSITES * NYEARS)   // 2,000,000 (s,y) elements
#define HDIM   64
#define NTILES (NE / 16)           // 125,000 tiles of 16 elements

// One wave handles 16 (s,y) elements per loop trip:
//   Hpre[64x16] = W_h[64x2] @ XY[2x16] + b_h   via 4x V_WMMA_F32_16X16X4_F32
//   h = elu(Hpre); two 64-dot reductions; sigmoids; fused store of psi and p.
__global__ __launch_bounds__(256) void occupancy_fused_kernel(
    const float* __restrict__ sxy,   // (NE, 2)
    const float* __restrict__ oxy,   // (NE, 2)
    const float* __restrict__ Wh,    // (64, 2)
    const float* __restrict__ bh,    // (64,)
    const float* __restrict__ Wpsi,  // (1, 64)
    const float* __restrict__ bpsi,  // (1,)
    const float* __restrict__ Wp,    // (1, 65)
    const float* __restrict__ bp,    // (1,)
    float* __restrict__ out)         // [0,NE): psi ; [NE,3*NE): p
{
    const int lane  = threadIdx.x & 31;
    const int half  = lane >> 4;     // 0: rows m%16<8 & K=0/1 ; 1: rows m%16>=8 & K=2/3
    const int col   = lane & 15;     // batch column / A-matrix row-in-chunk
    const int gwave = blockIdx.x * (blockDim.x >> 5) + (threadIdx.x >> 5);
    const int nwave = gridDim.x * (blockDim.x >> 5);

    // ---- per-wave constant state (L2-hot loads, done once) ----
    // A chunks: chunk c covers W_h rows [16c, 16c+16). K=2,3 lanes are zero.
    v2f A[4];
#pragma unroll
    for (int c = 0; c < 4; ++c) {
        float a0 = 0.0f, a1 = 0.0f;
        if (half == 0) {
            a0 = Wh[(16 * c + col) * 2 + 0];
            a1 = Wh[(16 * c + col) * 2 + 1];
        }
        A[c].x = a0; A[c].y = a1;
    }
    // Bias accumulators (C/D layout: vgpr r, lanes 0-15 -> M=r, lanes 16-31 -> M=r+8)
    // and per-lane slices of the two reduction weight vectors.
    v8f   Cb[4];
    float wps[4][8], whp[4][8];
#pragma unroll
    for (int c = 0; c < 4; ++c) {
#pragma unroll
        for (int r = 0; r < 8; ++r) {
            const int m = 16 * c + 8 * half + r;
            Cb[c][r]  = bh[m];
            wps[c][r] = Wpsi[m];   // W_psi[0, m]
            whp[c][r] = Wp[m];     // W_p[0, m]  (h-part)
        }
    }
    const float w_x  = Wp[HDIM];   // W_p[0, 64]
    const float b_p0 = bp[0];
    const float b_ps = bpsi[0];

    // ---- grid-stride over tiles of 16 elements ----
    for (int t = gwave; t < NTILES; t += nwave) {
        const int e = t * 16 + col;            // this lane's (s,y) element

        // prefetch next trip's streams (speculative, L2)
        if (t + nwave < NTILES) {
            __builtin_prefetch(sxy + 2 * (size_t)(t + nwave) * 16 + 2 * col, 0, 1);
            __builtin_prefetch(oxy + 2 * (size_t)(t + nwave) * 16 + 2 * col, 0, 1);
        }

        // B-matrix 4x16: lane n holds x_n (K=0) and y_n (K=1); upper half (K=2,3)=0
        const float2 xy = reinterpret_cast<const float2*>(sxy)[e];
        v2f B;
        B.x = half ? 0.0f : xy.x;
        B.y = half ? 0.0f : xy.y;

        // Hpre = W_h @ XY + b_h  (bias folded into C)
        v8f H[4];
#pragma unroll
        for (int c = 0; c < 4; ++c) {
            H[c] = __builtin_amdgcn_wmma_f32_16x16x4_f32(
                false, A[c], false, B, (short)0, Cb[c], false, false);
        }

        // ELU + the two 64-long dot products (32 rows live in this lane)
        float p1 = 0.0f, p2 = 0.0f;
#pragma unroll
        for (int c = 0; c < 4; ++c) {
#pragma unroll
            for (int r = 0; r < 8; ++r) {
                const float v  = H[c][r];
                const float ev = __expf(v) - 1.0f;
                const float h  = v > 0.0f ? v : ev;     // elu
                p1 = __builtin_fmaf(wps[c][r], h, p1);
                p2 = __builtin_fmaf(whp[c][r], h, p2);
            }
        }
        // fold the two half-row partials (lane n <-> lane n+16)
        p1 += __shfl_xor(p1, 16, 32);
        p2 += __shfl_xor(p2, 16, 32);

        // p_out[e, half] = sigmoid(h.w_hpart + oxy*w_x + b_p)
        const float ox     = oxy[2 * e + half];
        const float logitp = __builtin_fmaf(ox, w_x, p2 + b_p0);
        out[NE + 2 * e + half] = 1.0f / (1.0f + __expf(-logitp));

        // psi[e] = sigmoid(h.w_psi + b_psi)   (one lane per element)
        if (half == 0) {
            out[e] = 1.0f / (1.0f + __expf(-(p1 + b_ps)));
        }
    }
}

extern "C" void kernel_launch(void* const* d_in, const int* in_sizes, int n_in,
                              void* d_out, int out_size, void* d_ws, size_t ws_size,
                              hipStream_t stream) {
    const float* sxy  = (const float*)d_in[0];
    const float* oxy  = (const float*)d_in[1];
    // d_in[2] = p (zeros, unused by the reference math)
    const float* Wh   = (const float*)d_in[3];
    const float* bh   = (const float*)d_in[4];
    const float* Wpsi = (const float*)d_in[5];
    const float* bpsi = (const float*)d_in[6];
    const float* Wp   = (const float*)d_in[7];
    const float* bp   = (const float*)d_in[8];

    dim3 block(256);   // 8 wave32s
    dim3 grid(2048);   // 16384 waves -> ~7.6 tiles each (amortizes weight setup)
    hipLaunchKernelGGL(occupancy_fused_kernel, grid, block, 0, stream,
                       sxy, oxy, Wh, bh, Wpsi, bpsi, Wp, bp, (float*)d_out);
}